// DecoderMeanP2CPDistance_13237089206493
// MI455X (gfx1250) — compile-verified
//
#include <hip/hip_runtime.h>
#include <hip/hip_bf16.h>

typedef __attribute__((ext_vector_type(2))) float v2f;
typedef __attribute__((ext_vector_type(8))) float v8f;

#define NSAMP 128
#define NCOLS 256          // 2 * NSAMP
#define EPS_P2CP 1e-12f

#if __has_builtin(__builtin_amdgcn_global_load_async_to_lds_b32)
#define HAVE_ASYNC_LDS 1
typedef __attribute__((address_space(3))) int* lds_i32p;
// param1: generic int* (global source), param2: LDS int*, imm offset, cpol
#define ASYNC_CP_B32(dst, src) \
  __builtin_amdgcn_global_load_async_to_lds_b32((int*)(src), (lds_i32p)(dst), 0, 0)
#else
#define HAVE_ASYNC_LDS 0
#endif

// -------------------------------------------------------------------------
// Kernel 1: shapes[P x 256] = outputs[P x NC] @ W_dec[NC x 256] + b_dec
// One wave32 per 16x16 output tile, V_WMMA_F32_16X16X4_F32.
// K-loop: full K=4 steps with UNGUARDED loads (no EXEC divergence), plus a
// branchless clamped tail (statically present, not executed for NC=12).
//
// ISA VGPR layouts (CDNA5 ISA §7.12.2):
//   A 16x4 f32 : lanes 0-15 -> (K=0,K=1), lanes 16-31 -> (K=2,K=3), M = lane&15
//   B 4x16 f32 : mirrored, N = lane&15
//   C/D 16x16  : VGPR r, lane L -> M = r + 8*(L>=16), N = L&15
// -------------------------------------------------------------------------
__global__ __launch_bounds__(256) void decode_wmma_kernel(
    const float* __restrict__ outputs, const float* __restrict__ W_dec,
    const float* __restrict__ b_dec, float* __restrict__ shapes,
    int P, int NC)
{
  const int wave = threadIdx.x >> 5;
  const int lane = threadIdx.x & 31;
  const int tile = blockIdx.x * 8 + wave;
  const int NT   = NCOLS / 16;                 // 16 tiles along N
  const int mt   = tile / NT;
  const int nt   = tile % NT;
  const int m0   = mt * 16;
  const int n0   = nt * 16;
  const int half = lane >> 4;                  // K pair select / M half
  const int l    = lane & 15;

  if (m0 >= P) return;                         // wave-uniform, EXEC stays full

  const float* __restrict__ arow = outputs + (size_t)(m0 + l) * NC;
  const float* __restrict__ bcol = W_dec + n0 + l;

  v8f acc = {};
  const int KF = NC >> 2;                      // full K=4 steps (3 for NC=12)
  for (int kk = 0; kk < KF; ++kk) {
    const int kb = kk * 4 + half * 2;
    v2f a, b;
    a.x = arow[kb + 0];
    a.y = arow[kb + 1];
    b.x = bcol[(size_t)(kb + 0) * NCOLS];
    b.y = bcol[(size_t)(kb + 1) * NCOLS];
    // 8 args: (neg_a, A, neg_b, B, c_mod, C, reuse_a, reuse_b)
    acc = __builtin_amdgcn_wmma_f32_16x16x4_f32(
        false, a, false, b, (short)0, acc, false, false);
  }
  if (NC & 3) {                                // uniform scalar branch
    const int kb = KF * 4 + half * 2;
    const int k0 = (kb + 0 < NC) ? kb + 0 : 0; // clamp -> always in-bounds load
    const int k1 = (kb + 1 < NC) ? kb + 1 : 0;
    const float t0 = arow[k0];
    const float t1 = arow[k1];
    const float w0 = bcol[(size_t)k0 * NCOLS];
    const float w1 = bcol[(size_t)k1 * NCOLS];
    v2f a, b;
    a.x = (kb + 0 < NC) ? t0 : 0.0f;           // v_cndmask, no EXEC games
    a.y = (kb + 1 < NC) ? t1 : 0.0f;
    b.x = (kb + 0 < NC) ? w0 : 0.0f;
    b.y = (kb + 1 < NC) ? w1 : 0.0f;
    acc = __builtin_amdgcn_wmma_f32_16x16x4_f32(
        false, a, false, b, (short)0, acc, false, false);
  }

  const float bias = b_dec[n0 + l];
  #pragma unroll
  for (int r = 0; r < 8; ++r) {
    const int m = m0 + r + half * 8;
    shapes[(size_t)m * NCOLS + n0 + l] = acc[r] + bias;
  }
}

// -------------------------------------------------------------------------
// Kernel 2: per (b,s) pair, symmetric mean point-to-closest-point distance.
// Block = 128 threads. Points packed per-index as float4 (ux,uy,vx,vy) in
// LDS -> one ds_load_b128 per inner iteration. Staged via async-to-LDS
// (ASYNCcnt path) when the toolchain exposes the builtin.
// sqrt is monotone, so sqrt(min(d2)+eps) == min(sqrt(d2+eps)).
// -------------------------------------------------------------------------
__global__ __launch_bounds__(128) void p2cp_kernel(
    const float* __restrict__ shapes, const float* __restrict__ targets,
    float* __restrict__ pairv)
{
  __shared__ float4 pts[NSAMP];                // (ux, uy, vx, vy) per point
  __shared__ float wsum[4];

  const int p = blockIdx.x;
  const int t = threadIdx.x;
  const size_t base = (size_t)p * NCOLS;

#if HAVE_ASYNC_LDS
  ASYNC_CP_B32(&pts[t].x, shapes + base + t);
  ASYNC_CP_B32(&pts[t].y, shapes + base + NSAMP + t);
  ASYNC_CP_B32(&pts[t].z, targets + base + t);
  ASYNC_CP_B32(&pts[t].w, targets + base + NSAMP + t);
  asm volatile("s_wait_asynccnt 0" ::: "memory");
#else
  pts[t] = make_float4(shapes[base + t], shapes[base + NSAMP + t],
                       targets[base + t], targets[base + NSAMP + t]);
#endif
  __syncthreads();

  const float4 mine = pts[t];
  const float ptx = mine.x, pty = mine.y;      // u-point t (row side)
  const float qtx = mine.z, qty = mine.w;      // v-point t (col side)
  float rmin = 3.4e38f, cmin = 3.4e38f;

  #pragma unroll 4
  for (int j = 0; j < NSAMP; ++j) {
    const float4 pj = pts[j];                  // single b128 LDS load
    float dx = ptx - pj.z, dy = pty - pj.w;    // |u_t - v_j|^2
    rmin = fminf(rmin, dx * dx + dy * dy);
    float ex = pj.x - qtx, ey = pj.y - qty;    // |u_j - v_t|^2
    cmin = fminf(cmin, ex * ex + ey * ey);
  }

  float val = sqrtf(rmin + EPS_P2CP) + sqrtf(cmin + EPS_P2CP);

  // wave32 reduce, then across the 4 waves via LDS
  #pragma unroll
  for (int off = 16; off > 0; off >>= 1)
    val += __shfl_xor(val, off, 32);
  if ((t & 31) == 0) wsum[t >> 5] = val;
  __syncthreads();
  if (t == 0) {
    float s = wsum[0] + wsum[1] + wsum[2] + wsum[3];
    pairv[p] = 0.5f * s / (float)NSAMP;        // 0.5*(mean_u2v + mean_v2u)
  }
}

// -------------------------------------------------------------------------
// Kernel 3: deterministic tree-sum of per-pair values -> scalar mean.
// -------------------------------------------------------------------------
__global__ __launch_bounds__(256) void reduce_kernel(
    const float* __restrict__ pairv, float* __restrict__ out, int P)
{
  __shared__ float sm[256];
  float s = 0.0f;
  for (int i = threadIdx.x; i < P; i += 256) s += pairv[i];
  sm[threadIdx.x] = s;
  __syncthreads();
  #pragma unroll
  for (int off = 128; off > 0; off >>= 1) {
    if ((int)threadIdx.x < off) sm[threadIdx.x] += sm[threadIdx.x + off];
    __syncthreads();
  }
  if (threadIdx.x == 0) out[0] = sm[0] / (float)P;
}

extern "C" void kernel_launch(void* const* d_in, const int* in_sizes, int n_in,
                              void* d_out, int out_size, void* d_ws, size_t ws_size,
                              hipStream_t stream) {
  const float* outputs = (const float*)d_in[0];   // [bs, seq, NC]
  const float* targets = (const float*)d_in[1];   // [bs, seq, 1, 2, 128]
  const float* W_dec   = (const float*)d_in[2];   // [NC, 256]
  const float* b_dec   = (const float*)d_in[3];   // [256]
  float* out = (float*)d_out;

  const int P  = in_sizes[1] / NCOLS;             // bs*seq = 4096
  const int NC = in_sizes[2] / NCOLS;             // 12

  // workspace: shapes (P*256 f32 = 4 MB) + per-pair values (P f32)
  float* shapes = (float*)d_ws;
  float* pairv  = shapes + (size_t)P * NCOLS;

  const int tiles  = (P / 16) * (NCOLS / 16);     // 4096 tiles
  const int blocks = (tiles + 7) / 8;             // 8 waves/block

  decode_wmma_kernel<<<blocks, 256, 0, stream>>>(outputs, W_dec, b_dec, shapes, P, NC);
  p2cp_kernel<<<P, 128, 0, stream>>>(shapes, targets, pairv);
  reduce_kernel<<<1, 256, 0, stream>>>(pairv, out, P);
}